// RandomHybridRosenbrock_16535624090208
// MI455X (gfx1250) — compile-verified
//
#include <hip/hip_runtime.h>

typedef float v2f __attribute__((ext_vector_type(2)));
typedef float v8f __attribute__((ext_vector_type(8)));

#define ROWS        32767          // N2 - 1
#define ROWLEN      1024           // N1 - 1
#define TOTAL_ITEMS (ROWS * 256)   // 4 columns per item
#define NB1         4096
#define TPB1        256
#define TPB2        256            // 8 waves in pass 2
#define NWAVE2      (TPB2 / 32)
#define CHUNKS_PER_WAVE (NB1 / 64 / NWAVE2)   // 4096 partials / 64-per-chunk / 8 waves = 8

// ---------------- Pass 1: bandwidth-bound streaming partial reduction ----------------
__global__ __launch_bounds__(TPB1)
void rosen_pass1(const float* __restrict__ x, const float* __restrict__ b,
                 float* __restrict__ partials) {
    __shared__ float sdata[TPB1];
    float acc = 0.0f;
    const int stride = gridDim.x * blockDim.x;
    for (int wid = blockIdx.x * blockDim.x + threadIdx.x; wid < TOTAL_ITEMS; wid += stride) {
        const int r    = wid >> 8;            // row 0..32766
        const int c0   = (wid & 255) << 2;    // column chunk start 0..1020
        const int base = r * ROWLEN + c0;     // element index; 16B-aligned
        // x[base .. base+3] = { y[r][c0-1] (or junk if c0==0), y[r][c0], y[r][c0+1], y[r][c0+2] }
        const float4 v  = *reinterpret_cast<const float4*>(x + base);
        const float4 bv = *reinterpret_cast<const float4*>(b + base);
        const float  e  = x[base + 4];        // y[r][c0+3]
        // wraparound: yprev for c==0 is y[r][1023] = x[r*1024 + 1024]
        const float prev = (c0 == 0) ? x[base + ROWLEN] : v.x;
        float t;
        t = v.y - prev * prev;  acc = fmaf(bv.x, t * t, acc);
        t = v.z - v.y * v.y;    acc = fmaf(bv.y, t * t, acc);
        t = v.w - v.z * v.z;    acc = fmaf(bv.z, t * t, acc);
        if (c0 != ROWLEN - 4) {               // column 1023 is excluded
            t = e - v.w * v.w;  acc = fmaf(bv.w, t * t, acc);
        }
    }
    sdata[threadIdx.x] = acc;
    __syncthreads();
    #pragma unroll
    for (int s = TPB1 / 2; s > 0; s >>= 1) {
        if (threadIdx.x < s) sdata[threadIdx.x] += sdata[threadIdx.x + s];
        __syncthreads();
    }
    if (threadIdx.x == 0) partials[blockIdx.x] = sdata[0];
}

// ---------------- Pass 2: WMMA-based reduction of 4096 partials + finalize ----------------
// Uses V_WMMA_F32_16X16X4_F32 with A = ones(16x4): D[i][j] += sum_k B[k][j].
// Sum over all 256 D elements equals 16 * (sum of the 64 B values) regardless of
// the exact B VGPR layout, so the result is layout-independent.
__global__ __launch_bounds__(TPB2)
void rosen_pass2(const float* __restrict__ partials, const float* __restrict__ x,
                 const float* __restrict__ a, const float* __restrict__ mu,
                 float* __restrict__ out) {
    __shared__ float wsum[NWAVE2];
    const int wave = threadIdx.x >> 5;
    const int lane = threadIdx.x & 31;

    v2f ones = {1.0f, 1.0f};
    v8f c    = {0.f, 0.f, 0.f, 0.f, 0.f, 0.f, 0.f, 0.f};

    // Pre-issue all loads for this wave's chunks (64 floats per chunk, 2 per lane).
    v2f bmat[CHUNKS_PER_WAVE];
    const int chunk0 = wave * CHUNKS_PER_WAVE;
    #pragma unroll
    for (int i = 0; i < CHUNKS_PER_WAVE; ++i) {
        const float2 t = *reinterpret_cast<const float2*>(
            partials + (chunk0 + i) * 64 + 2 * lane);
        bmat[i].x = t.x;
        bmat[i].y = t.y;
    }
    // Accumulate column sums via the f32 matrix pipe (EXEC is all-1s: no divergence here).
    #pragma unroll
    for (int i = 0; i < CHUNKS_PER_WAVE; ++i) {
        c = __builtin_amdgcn_wmma_f32_16x16x4_f32(
                /*neg_a=*/false, ones, /*neg_b=*/false, bmat[i],
                /*c_mod=*/(short)0, c, /*reuse_a=*/false, /*reuse_b=*/false);
    }
    float s = c[0] + c[1] + c[2] + c[3] + c[4] + c[5] + c[6] + c[7];
    #pragma unroll
    for (int off = 16; off > 0; off >>= 1)
        s += __shfl_xor(s, off, 32);
    if (lane == 0) wsum[wave] = s;
    __syncthreads();

    if (threadIdx.x == 0) {
        float S = 0.0f;
        #pragma unroll
        for (int w = 0; w < NWAVE2; ++w) S += wsum[w];
        S *= (1.0f / 16.0f);                  // each B value was counted 16x by D
        const float d = x[0] - mu[0];
        out[0] = -a[0] * d * d - S;
    }
}

extern "C" void kernel_launch(void* const* d_in, const int* in_sizes, int n_in,
                              void* d_out, int out_size, void* d_ws, size_t ws_size,
                              hipStream_t stream) {
    const float* x  = (const float*)d_in[0];
    const float* a  = (const float*)d_in[1];
    const float* b  = (const float*)d_in[2];
    const float* mu = (const float*)d_in[3];
    float* out      = (float*)d_out;
    float* partials = (float*)d_ws;   // NB1 floats = 16 KB; fully rewritten each call

    rosen_pass1<<<NB1, TPB1, 0, stream>>>(x, b, partials);
    rosen_pass2<<<1, TPB2, 0, stream>>>(partials, x, a, mu, out);
}